// SkeletonMamba_15831249453206
// MI455X (gfx1250) — compile-verified
//
#include <hip/hip_runtime.h>
#include <math.h>

// ---- problem constants (from reference) ----
#define NDEPTH 4
#define EMB    192          // E
#define DIM    384          // DI
#define DXZ    768          // 2*DI
#define NST    16           // N (SSM state)
#define RRANK  12           // R
#define KCV    4            // conv kernel
#define NJ     42           // J
#define NT     16           // T
#define NB     2            // B
#define NCLS   1000         // NC
#define LSEQ   673          // 1 + T*J
#define MTOK   (NB*LSEQ)    // 1346 tokens
#define XDLD   48           // padded (R+2N)=44 -> 48 row stride
#define EPSR   1e-5f

typedef float v2f __attribute__((ext_vector_type(2)));
typedef float v8f __attribute__((ext_vector_type(8)));

__device__ __forceinline__ float sigmoidf_(float x) { return 1.f / (1.f + __expf(-x)); }
__device__ __forceinline__ float siluf_(float x)    { return x * sigmoidf_(x); }
__device__ __forceinline__ float softplusf_(float x){ return x > 20.f ? x : __logf(1.f + __expf(x)); }

// ---------------- token embedding: joint proj + cls + positional ----------------
__global__ void k_embed(const float* __restrict__ x, const float* __restrict__ jw,
                        const float* __restrict__ jb, const float* __restrict__ cls,
                        const float* __restrict__ sp, const float* __restrict__ tp,
                        float* __restrict__ h) {
  int idx = blockIdx.x * blockDim.x + threadIdx.x;
  if (idx >= MTOK * EMB) return;
  int e  = idx % EMB;
  int tk = idx / EMB;          // b*LSEQ + l
  int l  = tk % LSEQ;
  int b  = tk / LSEQ;
  float v;
  if (l == 0) {
    v = cls[e] + sp[e];        // spatial_pos[0,0,e]
  } else {
    int t = (l - 1) / NJ, j = (l - 1) % NJ;
    const float* xp = x + (((size_t)(b * NT + t) * NJ + j) * 3);
    v = jb[e] + xp[0] * jw[e * 3 + 0] + xp[1] * jw[e * 3 + 1] + xp[2] * jw[e * 3 + 2]
        + sp[(j + 1) * EMB + e] + tp[t * EMB + e];
  }
  h[idx] = v;
}

// ---------------- residual accumulate + RMSNorm (block per token, 192 thr) ------
__global__ void k_resnorm(const float* __restrict__ h, float* __restrict__ res,
                          float* __restrict__ hn, const float* __restrict__ w, int first) {
  int t = blockIdx.x, e = threadIdx.x;
  __shared__ float sd[EMB];
  size_t i = (size_t)t * EMB + e;
  float v = h[i] + (first ? 0.f : res[i]);
  res[i] = v;
  sd[e] = v * v;
  __syncthreads();
  if (e < 64) sd[e] += sd[e + 64] + sd[e + 128];
  __syncthreads();
  for (int s = 32; s > 0; s >>= 1) { if (e < s) sd[e] += sd[e + s]; __syncthreads(); }
  float scale = rsqrtf(sd[0] / (float)EMB + EPSR);
  hn[i] = v * scale * w[e];
}

// ---------------- fp32 WMMA GEMM: C[M,N] = A[M,K] * W[N,K]^T --------------------
// one wave32 per 16x16 tile; V_WMMA_F32_16X16X4_F32, K stepped by 4.
// Out-of-range rows/cols are CLAMPED (not zeroed): a garbage A-row m only
// pollutes output row m, a garbage W-col n only output col n, and those are
// never stored. This keeps the inner loop branch-free: 2x b64 load + wmma.
__global__ void k_gemm_wmma(const float* __restrict__ A, const float* __restrict__ W,
                            float* __restrict__ C, int Mm, int Nn, int Kk,
                            int lda, int ldw, int ldc) {
  int lane = threadIdx.x;
  int m0 = blockIdx.x * 16, n0 = blockIdx.y * 16;
  int half = lane >> 4;          // 0: K+{0,1}, 1: K+{2,3}
  int kk = half << 1;
  int row = m0 + (lane & 15);    // A fragment: M = lane&15
  int col = n0 + (lane & 15);    // B fragment: N = lane&15
  int rowc = row < Mm ? row : Mm - 1;
  int colc = col < Nn ? col : Nn - 1;
  const float* ap = A + (size_t)rowc * lda + kk;
  const float* wp = W + (size_t)colc * ldw + kk;
  v8f acc = {0.f, 0.f, 0.f, 0.f, 0.f, 0.f, 0.f, 0.f};
#pragma unroll 8
  for (int k0 = 0; k0 < Kk; k0 += 4) {
    v2f a  = *(const v2f*)(ap + k0);   // global_load_b64
    v2f bm = *(const v2f*)(wp + k0);   // global_load_b64
    acc = __builtin_amdgcn_wmma_f32_16x16x4_f32(false, a, false, bm, (short)0, acc,
                                                false, false);
  }
  if (col < Nn) {
#pragma unroll
    for (int r = 0; r < 8; r++) {
      int crow = m0 + half * 8 + r;     // C/D layout: lanes>=16 hold M=8..15
      if (crow < Mm) C[(size_t)crow * ldc + col] = acc[r];
    }
  }
}

// ---------------- causal depthwise conv (K=4) + SiLU, both directions -----------
__global__ void k_conv_silu(const float* __restrict__ xz, const float* __restrict__ cw,
                            const float* __restrict__ cb, float* __restrict__ xcf,
                            float* __restrict__ xcb) {
  int idx = blockIdx.x * blockDim.x + threadIdx.x;
  if (idx >= MTOK * DIM) return;
  int dir = blockIdx.y;
  int d = idx % DIM;
  int p = (idx / DIM) % LSEQ;
  int b = idx / (DIM * LSEQ);
  const float* cwd = cw + ((size_t)dir * DIM + d) * KCV;
  float acc = cb[dir * DIM + d];
#pragma unroll
  for (int k = 0; k < KCV; k++) {
    int q = p - (KCV - 1) + k;
    if (q >= 0) {
      int ls = dir ? (LSEQ - 1 - q) : q;
      acc += xz[(size_t)(b * LSEQ + ls) * DXZ + d] * cwd[k];
    }
  }
  float* outp = dir ? xcb : xcf;
  outp[(size_t)(b * LSEQ + p) * DIM + d] = siluf_(acc);
}

// ---------------- selective scan: thread per channel d, 16 states in regs -------
__global__ void k_scan(const float* __restrict__ xdbl, const float* __restrict__ xc,
                       const float* __restrict__ xz, const float* __restrict__ dtw,
                       const float* __restrict__ dtb, const float* __restrict__ Alog,
                       const float* __restrict__ Dp, float* __restrict__ y, int dir) {
  int b = blockIdx.x;
  int d = threadIdx.x;
  float wdt[RRANK];
#pragma unroll
  for (int r = 0; r < RRANK; r++) wdt[r] = dtw[(size_t)d * RRANK + r];
  float negA[NST], hs[NST];
#pragma unroll
  for (int n = 0; n < NST; n++) { negA[n] = -__expf(Alog[(size_t)d * NST + n]); hs[n] = 0.f; }
  float dtbias = dtb[d];
  float Dv = Dp[d];
  for (int l = 0; l < LSEQ; l++) {
    const float* xr = xdbl + (size_t)(b * LSEQ + l) * XDLD;
    float dtraw = dtbias;
#pragma unroll
    for (int r = 0; r < RRANK; r++) dtraw += xr[r] * wdt[r];
    float dt = softplusf_(dtraw);
    float xcv = xc[(size_t)(b * LSEQ + l) * DIM + d];
    float yv = 0.f;
#pragma unroll
    for (int n = 0; n < NST; n++) {
      float dA = __expf(dt * negA[n]);
      hs[n] = dA * hs[n] + dt * xr[RRANK + n] * xcv;
      yv += hs[n] * xr[RRANK + NST + n];
    }
    yv += Dv * xcv;
    int zl = dir ? (LSEQ - 1 - l) : l;
    float zv = xz[(size_t)(b * LSEQ + zl) * DXZ + DIM + d];
    y[(size_t)(b * LSEQ + l) * DIM + d] = yv * siluf_(zv);
  }
}

// ---------------- combine fwd + reversed-bwd ------------------------------------
__global__ void k_combine(const float* __restrict__ yf, const float* __restrict__ yb,
                          float* __restrict__ yc) {
  int idx = blockIdx.x * blockDim.x + threadIdx.x;
  if (idx >= MTOK * DIM) return;
  int d = idx % DIM;
  int l = (idx / DIM) % LSEQ;
  int b = idx / (DIM * LSEQ);
  yc[idx] = yf[idx] + yb[(size_t)(b * LSEQ + (LSEQ - 1 - l)) * DIM + d];
}

// ---------------- final RMSNorm of token 0 only ---------------------------------
__global__ void k_final_norm(const float* __restrict__ h, const float* __restrict__ res,
                             const float* __restrict__ w, float* __restrict__ hf0) {
  int b = blockIdx.x, e = threadIdx.x;
  __shared__ float sd[EMB];
  size_t i = (size_t)(b * LSEQ) * EMB + e;
  float v = h[i] + res[i];
  sd[e] = v * v;
  __syncthreads();
  if (e < 64) sd[e] += sd[e + 64] + sd[e + 128];
  __syncthreads();
  for (int s = 32; s > 0; s >>= 1) { if (e < s) sd[e] += sd[e + s]; __syncthreads(); }
  float scale = rsqrtf(sd[0] / (float)EMB + EPSR);
  hf0[b * EMB + e] = v * scale * w[e];
}

// ---------------- classifier head (2x1000, thread per output) -------------------
__global__ void k_head(const float* __restrict__ hf0, const float* __restrict__ hw,
                       const float* __restrict__ hb, float* __restrict__ out) {
  int idx = blockIdx.x * blockDim.x + threadIdx.x;
  if (idx >= NB * NCLS) return;
  int c = idx % NCLS, b = idx / NCLS;
  const float* hv = hf0 + b * EMB;
  const float* wr = hw + (size_t)c * EMB;
  float acc = hb[c];
#pragma unroll 8
  for (int e = 0; e < EMB; e++) acc += hv[e] * wr[e];
  out[idx] = acc;
}

extern "C" void kernel_launch(void* const* d_in, const int* in_sizes, int n_in,
                              void* d_out, int out_size, void* d_ws, size_t ws_size,
                              hipStream_t stream) {
  (void)in_sizes; (void)n_in; (void)out_size; (void)ws_size;
  const float* x    = (const float*)d_in[0];
  const float* jw   = (const float*)d_in[1];
  const float* jb   = (const float*)d_in[2];
  const float* cls  = (const float*)d_in[3];
  const float* sp   = (const float*)d_in[4];
  const float* tp   = (const float*)d_in[5];
  const float* ipw  = (const float*)d_in[6];
  const float* cw   = (const float*)d_in[7];
  const float* cb   = (const float*)d_in[8];
  const float* xpw  = (const float*)d_in[9];
  const float* dtw  = (const float*)d_in[10];
  const float* dtb  = (const float*)d_in[11];
  const float* Alog = (const float*)d_in[12];
  const float* Dp   = (const float*)d_in[13];
  const float* ow   = (const float*)d_in[14];
  const float* nw   = (const float*)d_in[15];
  const float* nfw  = (const float*)d_in[16];
  const float* hw   = (const float*)d_in[17];
  const float* hb   = (const float*)d_in[18];
  float* out = (float*)d_out;
  float* ws  = (float*)d_ws;

  // workspace layout (floats), ~4.53M floats total (~18.1 MB)
  float* h   = ws;
  float* res = h   + (size_t)MTOK * EMB;
  float* hn  = res + (size_t)MTOK * EMB;
  float* xz  = hn  + (size_t)MTOK * EMB;
  float* xcf = xz  + (size_t)MTOK * DXZ;
  float* xcb = xcf + (size_t)MTOK * DIM;
  float* xdf = xcb + (size_t)MTOK * DIM;
  float* xdb = xdf + (size_t)MTOK * XDLD;
  float* yf  = xdb + (size_t)MTOK * XDLD;
  float* yb  = yf  + (size_t)MTOK * DIM;
  float* yc  = yb  + (size_t)MTOK * DIM;
  float* hf0 = yc  + (size_t)MTOK * DIM;

  const int MT = (MTOK + 15) / 16;   // 85 M-tiles
  const int RN = RRANK + 2 * NST;    // 44
  dim3 wave(32);

  k_embed<<<(MTOK * EMB + 255) / 256, 256, 0, stream>>>(x, jw, jb, cls, sp, tp, h);

  for (int i = 0; i < NDEPTH; i++) {
    k_resnorm<<<MTOK, EMB, 0, stream>>>(h, res, hn, nw + (size_t)i * EMB, i == 0);

    // in_proj: [1346,192] x [768,192]^T -> xz [1346,768]
    k_gemm_wmma<<<dim3(MT, DXZ / 16), wave, 0, stream>>>(
        hn, ipw + (size_t)i * DXZ * EMB, xz, MTOK, DXZ, EMB, EMB, EMB, DXZ);

    k_conv_silu<<<dim3((MTOK * DIM + 255) / 256, 2), 256, 0, stream>>>(
        xz, cw + (size_t)i * 2 * DIM * KCV, cb + (size_t)i * 2 * DIM, xcf, xcb);

    // xproj both dirs: [1346,384] x [44,384]^T -> [1346,44] (stride 48)
    k_gemm_wmma<<<dim3(MT, 3), wave, 0, stream>>>(
        xcf, xpw + (size_t)(i * 2 + 0) * RN * DIM, xdf, MTOK, RN, DIM, DIM, DIM, XDLD);
    k_gemm_wmma<<<dim3(MT, 3), wave, 0, stream>>>(
        xcb, xpw + (size_t)(i * 2 + 1) * RN * DIM, xdb, MTOK, RN, DIM, DIM, DIM, XDLD);

    k_scan<<<NB, DIM, 0, stream>>>(xdf, xcf, xz,
        dtw + (size_t)(i * 2 + 0) * DIM * RRANK, dtb + (size_t)(i * 2 + 0) * DIM,
        Alog + (size_t)(i * 2 + 0) * DIM * NST, Dp + (size_t)(i * 2 + 0) * DIM, yf, 0);
    k_scan<<<NB, DIM, 0, stream>>>(xdb, xcb, xz,
        dtw + (size_t)(i * 2 + 1) * DIM * RRANK, dtb + (size_t)(i * 2 + 1) * DIM,
        Alog + (size_t)(i * 2 + 1) * DIM * NST, Dp + (size_t)(i * 2 + 1) * DIM, yb, 1);

    k_combine<<<(MTOK * DIM + 255) / 256, 256, 0, stream>>>(yf, yb, yc);

    // out_proj: [1346,384] x [192,384]^T -> h [1346,192]
    k_gemm_wmma<<<dim3(MT, EMB / 16), wave, 0, stream>>>(
        yc, ow + (size_t)i * EMB * DIM, h, MTOK, EMB, DIM, DIM, DIM, EMB);
  }

  k_final_norm<<<NB, EMB, 0, stream>>>(h, res, nfw, hf0);
  k_head<<<(NB * NCLS + 255) / 256, 256, 0, stream>>>(hf0, hw, hb, out);
}